// PostProcess_2465311228507
// MI455X (gfx1250) — compile-verified
//
#include <hip/hip_runtime.h>
#include <stdint.h>

// Problem constants (from reference)
#define Bn      64
#define Qn      2000
#define Cn      365
#define QC      (Qn * Cn)        // 730000 scores per image
#define QC4     (QC / 4)         // 182500 float4 quads (exact)
#define Kk      100
#define NBINS   4096             // bins = float_bits >> 19 (scores in (0,1) -> <= 2032)
#define BSHIFT  19
#define CAND    2048             // candidate buffer (expected ~100-1000 survivors)
#define SLICES  8                // blocks per image in histogram pass
#define THREADS 1024             // 32 wave32 waves per workgroup

// ---- CDNA5 async global->LDS copy (gfx1250, ASYNCcnt-tracked) ----
__device__ __forceinline__ void async_b128_to_lds(const void* gptr, void* ldsptr) {
    unsigned lds32 = (unsigned)(uintptr_t)ldsptr;   // low 32 bits of LDS-aperture address
    asm volatile("global_load_async_to_lds_b128 %0, %1, off"
                 :: "v"(lds32), "v"(gptr) : "memory");
}
__device__ __forceinline__ void wait_async_zero() {
    asm volatile("s_wait_asynccnt 0" ::: "memory");
}

__global__ void pp_zero(unsigned* __restrict__ p, int n) {
    int i = blockIdx.x * blockDim.x + threadIdx.x;
    if (i < n) p[i] = 0u;
}

// Pass 1: per-image histogram of score float-bit bins. grid = (SLICES, B)
__global__ __launch_bounds__(THREADS)
void pp_hist(const float* __restrict__ logits, const float* __restrict__ obj,
             unsigned* __restrict__ ghist) {
    const int b   = blockIdx.y;
    const int s   = blockIdx.x;
    const int tid = threadIdx.x;
    __shared__ float    s_objw[Qn];
    __shared__ unsigned s_hist[NBINS];

    if (tid < Qn / 4)   // 500 lanes stage 8 KB of obj via async DMA
        async_b128_to_lds(obj + (size_t)b * Qn + tid * 4, &s_objw[tid * 4]);
    for (int i = tid; i < NBINS; i += THREADS) s_hist[i] = 0u;
    wait_async_zero();
    __syncthreads();
    for (int q = tid; q < Qn; q += THREADS) s_objw[q] = __expf(-s_objw[q]);
    __syncthreads();

    const float4* Lq  = reinterpret_cast<const float4*>(logits + (size_t)b * QC);
    const int per = (QC4 + SLICES - 1) / SLICES;
    const int q0  = s * per;
    const int q1  = min(q0 + per, QC4);
    for (int qi = q0 + tid; qi < q1; qi += THREADS) {
        __builtin_prefetch((const char*)(Lq + qi) + 32768, 0, 0);  // global_prefetch_b8
        float4 v = Lq[qi];
        const int base = qi * 4;
        #pragma unroll
        for (int j = 0; j < 4; ++j) {
            float x = (j == 0) ? v.x : (j == 1) ? v.y : (j == 2) ? v.z : v.w;
            int idx = base + j;
            int q   = idx / Cn;
            int c   = idx - q * Cn;
            float sc = (c == 0) ? 0.0f : s_objw[q] * (1.0f / (1.0f + __expf(-x)));
            atomicAdd(&s_hist[__float_as_uint(sc) >> BSHIFT], 1u);
        }
    }
    __syncthreads();
    unsigned* gh = ghist + (size_t)b * NBINS;
    for (int i = tid; i < NBINS; i += THREADS) {
        unsigned v = s_hist[i];
        if (v) atomicAdd(&gh[i], v);
    }
}

// Pass 2: threshold, compare-only rescan (L2-resident), bitonic top-K, output.
__global__ __launch_bounds__(THREADS)
void pp_select(const float* __restrict__ logits, const float* __restrict__ obj,
               const float* __restrict__ boxes,  const float* __restrict__ tsize,
               const unsigned* __restrict__ ghist, float* __restrict__ out) {
    const int b   = blockIdx.x;
    const int tid = threadIdx.x;
    __shared__ float s_objw[Qn];
    __shared__ float s_xmin[Qn];
    __shared__ unsigned long long s_cand[CAND];
    __shared__ unsigned s_cnt;
    __shared__ unsigned s_lbbits;

    if (tid < Qn / 4)
        async_b128_to_lds(obj + (size_t)b * Qn + tid * 4, &s_objw[tid * 4]);
    for (int i = tid; i < CAND; i += THREADS) s_cand[i] = 0ull;
    wait_async_zero();
    __syncthreads();
    for (int q = tid; q < Qn; q += THREADS) s_objw[q] = __expf(-s_objw[q]);
    if (tid == 0) {
        const unsigned* gh = ghist + (size_t)b * NBINS;
        unsigned cum = 0; int T = 0;
        for (int i = 2047; i >= 0; --i) {           // expected to stop in a few steps
            cum += gh[i];
            if (cum >= Kk) { T = i; break; }
        }
        s_lbbits = (unsigned)T << BSHIFT;
        s_cnt = 0u;
    }
    __syncthreads();

    // Per-query logit threshold: objw[q]*sigmoid(x) >= lb  <=>  x >= logit(lb/objw[q])
    const float lb = __uint_as_float(s_lbbits);
    for (int q = tid; q < Qn; q += THREADS) {
        float r = lb / s_objw[q];
        s_xmin[q] = (r >= 0.999999f) ? __builtin_inff()
                                     : (__logf(r / (1.0f - r)) - 1e-3f);
    }
    __syncthreads();

    const float4* Lq = reinterpret_cast<const float4*>(logits + (size_t)b * QC);
    for (int qi = tid; qi < QC4; qi += THREADS) {
        float4 v = Lq[qi];
        const int base = qi * 4;
        #pragma unroll
        for (int j = 0; j < 4; ++j) {
            float x = (j == 0) ? v.x : (j == 1) ? v.y : (j == 2) ? v.z : v.w;
            int idx = base + j;
            int q   = idx / Cn;
            int c   = idx - q * Cn;
            if (c != 0 && x >= s_xmin[q]) {
                float sc = s_objw[q] * (1.0f / (1.0f + __expf(-x)));
                unsigned pos = atomicAdd(&s_cnt, 1u);
                if (pos < CAND) {
                    // key: score bits high, ~idx low => descending sort breaks ties by lower idx
                    s_cand[pos] = ((unsigned long long)__float_as_uint(sc) << 32)
                                | (unsigned)(0xFFFFFFFFu ^ (unsigned)idx);
                }
            }
        }
    }
    __syncthreads();

    // Bitonic sort, descending, CAND = 2048 elements with 1024 threads
    for (int k = 2; k <= CAND; k <<= 1) {
        for (int j = k >> 1; j > 0; j >>= 1) {
            for (int i = tid; i < CAND; i += THREADS) {
                int l = i ^ j;
                if (l > i) {
                    unsigned long long a = s_cand[i], c2 = s_cand[l];
                    bool descSeg = ((i & k) == 0);
                    if (descSeg ? (a < c2) : (a > c2)) { s_cand[i] = c2; s_cand[l] = a; }
                }
            }
            __syncthreads();
        }
    }

    if (tid < Kk) {
        unsigned long long key = s_cand[tid];
        float sc = 0.0f; int lab = 0;
        float b0 = 0.f, b1 = 0.f, b2 = 0.f, b3 = 0.f;
        if (key) {
            sc = __uint_as_float((unsigned)(key >> 32));
            unsigned idx = 0xFFFFFFFFu ^ (unsigned)key;
            int q = (int)(idx / Cn);
            lab   = (int)(idx - (unsigned)q * Cn);
            const float* bp = boxes + ((size_t)b * Qn + q) * 4;
            float cx = bp[0], cy = bp[1], w = bp[2], h = bp[3];
            float ih = tsize[b * 2 + 0], iw = tsize[b * 2 + 1];
            b0 = (cx - 0.5f * w) * iw;
            b1 = (cy - 0.5f * h) * ih;
            b2 = (cx + 0.5f * w) * iw;
            b3 = (cy + 0.5f * h) * ih;
        }
        out[(size_t)b * Kk + tid] = sc;                                   // scores
        out[(size_t)Bn * Kk + (size_t)b * Kk + tid] = (float)lab;         // labels
        float* ob = out + (size_t)2 * Bn * Kk + ((size_t)b * Kk + tid) * 4;
        ob[0] = b0; ob[1] = b1; ob[2] = b2; ob[3] = b3;                   // boxes
    }
}

extern "C" void kernel_launch(void* const* d_in, const int* in_sizes, int n_in,
                              void* d_out, int out_size, void* d_ws, size_t ws_size,
                              hipStream_t stream) {
    (void)in_sizes; (void)n_in; (void)out_size; (void)ws_size;
    const float* logits = (const float*)d_in[0];
    const float* obj    = (const float*)d_in[1];
    const float* boxes  = (const float*)d_in[2];
    const float* tsize  = (const float*)d_in[3];
    float* out      = (float*)d_out;
    unsigned* ghist = (unsigned*)d_ws;      // needs B*NBINS*4 = 1 MiB of scratch

    const int nh = Bn * NBINS;
    pp_zero<<<(nh + 255) / 256, 256, 0, stream>>>(ghist, nh);
    pp_hist<<<dim3(SLICES, Bn), THREADS, 0, stream>>>(logits, obj, ghist);
    pp_select<<<Bn, THREADS, 0, stream>>>(logits, obj, boxes, tsize, ghist, out);
}